// ScaledDotProductAttention_77275051590120
// MI455X (gfx1250) — compile-verified
//
#include <hip/hip_runtime.h>
#include <math.h>
#include <stdint.h>

// ---- problem constants (from reference) ----
#define S_LEN 2048
#define DKDIM 64
#define BH    32               // B*H
#define QT_PER_BH (S_LEN/16)   // 128 tiles of 16
#define NEGV  (-1e9f)
#define WAVES 4                // waves per block (wave32)
#define NTHREADS (WAVES*32)

typedef __attribute__((ext_vector_type(2))) float v2f;
typedef __attribute__((ext_vector_type(8))) float v8f;

// Stage one contiguous 4KB tile (16 rows x 64 f32) global -> LDS with the
// CDNA5 async DMA path. 128 threads x 32B each, two b128 transfers per thread.
// GV addressing: vdst = LDS byte address, vaddr = 64-bit global address,
// INST_OFFSET is added on both sides (CDNA5 ISA).
__device__ __forceinline__ void async_stage_4k(const float* __restrict__ gsrc,
                                               float* lds_dst, int tid)
{
    uint32_t l = (uint32_t)(uintptr_t)lds_dst + (uint32_t)tid * 32u;
    const char* g = (const char*)gsrc + (size_t)tid * 32u;
    asm volatile(
        "global_load_async_to_lds_b128 %0, %1, off\n\t"
        "global_load_async_to_lds_b128 %0, %1, off offset:16"
        :: "v"(l), "v"(g) : "memory");
}

// Wait for this wave's async DMA (ASYNCcnt), then block barrier: after this,
// every wave's staged data is visible in LDS.
__device__ __forceinline__ void wait_async_and_barrier()
{
    asm volatile("s_wait_asynccnt 0x0" ::: "memory");
    __syncthreads();
}

__global__ __launch_bounds__(NTHREADS)
void sdpa_wmma_f32_kernel(const float* __restrict__ Q,
                          const float* __restrict__ K,
                          const float* __restrict__ V,
                          const unsigned char* __restrict__ mask,  // jnp bool = 1 byte
                          float* __restrict__ scores_out,
                          float* __restrict__ ctx_out,
                          float* __restrict__ attn_out)
{
    __shared__ float kvbuf[2][16 * DKDIM];   // double-buffered K/V tile (2 x 4KB)
    __shared__ float s_m [WAVES][16];
    __shared__ float s_il[WAVES][16];

    const int tid  = threadIdx.x;
    const int wave = tid >> 5;
    const int lane = tid & 31;
    const int half = lane >> 4;      // K-pair / M-half selector
    const int ln   = lane & 15;      // N (and A-row) index

    const int tile  = blockIdx.x * WAVES + wave;   // 0..4095 (block shares bh)
    const int bh    = tile / QT_PER_BH;
    const int qbase = (tile % QT_PER_BH) * 16;

    const float scale = 0.125f;      // 1/sqrt(64)

    // ---------------- preload Q tile as A-fragments ----------------
    // A[lane][v] = Q[qbase + ln][4*ks + 2*half + v]
    const float* qrow = Q + ((size_t)bh * S_LEN + qbase + ln) * DKDIM + 2 * half;
    v2f qf[16];
#pragma unroll
    for (int ks = 0; ks < 16; ++ks) {
        qf[ks].x = qrow[4 * ks + 0];
        qf[ks].y = qrow[4 * ks + 1];
    }

    // ---------------- pass 1: scores + online softmax stats ----------------
    float mrun[8], lrun[8];
#pragma unroll
    for (int r = 0; r < 8; ++r) { mrun[r] = -3.0e38f; lrun[r] = 0.0f; }

    const float* kbh = K + (size_t)bh * S_LEN * DKDIM;     // tiles are contiguous 4KB
    const unsigned char* mbase = mask + ((size_t)bh * S_LEN + qbase) * S_LEN;
    float* sbase = scores_out + ((size_t)bh * S_LEN + qbase) * S_LEN;

    // prologue: stage K tile 0
    async_stage_4k(kbh, kvbuf[0], tid);
    wait_async_and_barrier();

    for (int kt = 0; kt < QT_PER_BH; ++kt) {
        const int kbase = kt * 16;

        // issue DMA for next K tile into the other buffer (overlaps compute)
        if (kt + 1 < QT_PER_BH)
            async_stage_4k(kbh + (size_t)(kbase + 16) * DKDIM,
                           kvbuf[(kt + 1) & 1], tid);

        // prefetch next mask tile row for this lane
        __builtin_prefetch(mbase + (size_t)ln * S_LEN + kbase + 16, 0, 0);

        // B[lane][v] = K[kbase + ln][4*ks + 2*half + v]  (from LDS)
        const float* kl = &kvbuf[kt & 1][ln * DKDIM + 2 * half];
        v8f c = {};
#pragma unroll
        for (int ks = 0; ks < 16; ++ks) {
            v2f kf;
            kf.x = kl[4 * ks + 0];
            kf.y = kl[4 * ks + 1];
            c = __builtin_amdgcn_wmma_f32_16x16x4_f32(false, qf[ks], false, kf,
                                                      (short)0, c, false, false);
        }

        // per-element: scale, mask, store scores, update running max/sum
#pragma unroll
        for (int r = 0; r < 8; ++r) {
            const int mrow = r + 8 * half;                 // C/D: M = r + 8*(lane/16)
            const size_t idx = (size_t)mrow * S_LEN + kbase + ln;  // N = lane%16
            float s = c[r] * scale;
            if (mbase[idx]) s = NEGV;
            sbase[idx] = s;

            float t = s;                                   // row max over 16 N-lanes
            t = fmaxf(t, __shfl_xor(t, 1));
            t = fmaxf(t, __shfl_xor(t, 2));
            t = fmaxf(t, __shfl_xor(t, 4));
            t = fmaxf(t, __shfl_xor(t, 8));
            const float mnew = fmaxf(mrun[r], t);

            float p = __expf(s - mnew);                    // row sum of exp
            p += __shfl_xor(p, 1);
            p += __shfl_xor(p, 2);
            p += __shfl_xor(p, 4);
            p += __shfl_xor(p, 8);
            lrun[r] = lrun[r] * __expf(mrun[r] - mnew) + p;
            mrun[r] = mnew;
        }

        wait_async_and_barrier();   // next tile staged; old buffer free to reuse
    }

    // redistribute stats: (r,half) layout  ->  per-q-row layout, via LDS
    if (ln == 0) {
#pragma unroll
        for (int r = 0; r < 8; ++r) {
            s_m [wave][8 * half + r] = mrun[r];
            s_il[wave][8 * half + r] = 1.0f / lrun[r];
        }
    }
    __syncthreads();
    const float m_q  = s_m [wave][ln];
    const float il_q = s_il[wave][ln];

    // ---------------- pass 2: attn + context ----------------
    const float* srow  = sbase + (size_t)ln * S_LEN;                          // q = qbase+ln
    float*       arow  = attn_out + ((size_t)bh * S_LEN + qbase + ln) * S_LEN;
    const float* vbh   = V + (size_t)bh * S_LEN * DKDIM;

    v8f acc[4] = {v8f{}, v8f{}, v8f{}, v8f{}};   // 16 x 64 context (4 N-chunks)

    // prologue: stage V tile 0
    async_stage_4k(vbh, kvbuf[0], tid);
    wait_async_and_barrier();

    for (int kt = 0; kt < QT_PER_BH; ++kt) {
        const int kbase = kt * 16;

        if (kt + 1 < QT_PER_BH)
            async_stage_4k(vbh + (size_t)(kbase + 16) * DKDIM,
                           kvbuf[(kt + 1) & 1], tid);

        v2f af[4];
#pragma unroll
        for (int ks = 0; ks < 4; ++ks) {
            const int k0 = kbase + 4 * ks + 2 * half;
            v2f p;
            p.x = __expf(srow[k0 + 0] - m_q) * il_q;       // scores hot in L2
            p.y = __expf(srow[k0 + 1] - m_q) * il_q;
            *(v2f*)(arow + k0) = p;                        // 8B-aligned (k0 even)
            af[ks] = p;                                    // A[lane][v] = P[q][k0+v]
        }

        const float* vl = kvbuf[kt & 1];
#pragma unroll
        for (int ks = 0; ks < 4; ++ks) {
            const int row = 4 * ks + 2 * half;             // row within the 16-row V tile
#pragma unroll
            for (int ch = 0; ch < 4; ++ch) {
                v2f bf;                                    // B[lane][v] = V[row+v][16*ch+ln]
                bf.x = vl[(row + 0) * DKDIM + 16 * ch + ln];
                bf.y = vl[(row + 1) * DKDIM + 16 * ch + ln];
                acc[ch] = __builtin_amdgcn_wmma_f32_16x16x4_f32(false, af[ks], false, bf,
                                                                (short)0, acc[ch], false, false);
            }
        }

        wait_async_and_barrier();
    }

    // context store: already normalized (P carried 1/l)
    float* cbase = ctx_out + ((size_t)bh * S_LEN + qbase) * DKDIM;
#pragma unroll
    for (int ch = 0; ch < 4; ++ch)
#pragma unroll
        for (int r = 0; r < 8; ++r)
            cbase[(size_t)(r + 8 * half) * DKDIM + 16 * ch + ln] = acc[ch][r];
}

extern "C" void kernel_launch(void* const* d_in, const int* in_sizes, int n_in,
                              void* d_out, int out_size, void* d_ws, size_t ws_size,
                              hipStream_t stream) {
    (void)in_sizes; (void)n_in; (void)out_size; (void)d_ws; (void)ws_size;
    const float* Q = (const float*)d_in[0];
    const float* K = (const float*)d_in[1];
    const float* V = (const float*)d_in[2];
    const unsigned char* mask = (const unsigned char*)d_in[3];  // bool array

    float* out    = (float*)d_out;
    float* scores = out;                                             // [B,H,S,S]
    float* ctx    = scores + (size_t)BH * S_LEN * S_LEN;             // [B,H,S,DK]
    float* attn   = ctx    + (size_t)BH * S_LEN * DKDIM;             // [B,H,S,S]

    const int nblocks = (BH * QT_PER_BH) / WAVES;                    // 1024
    sdpa_wmma_f32_kernel<<<nblocks, NTHREADS, 0, stream>>>(
        Q, K, V, mask, scores, ctx, attn);
}